// LSTM_40003325395630
// MI455X (gfx1250) — compile-verified
//
#include <hip/hip_runtime.h>
#include <cstdint>
#include <cstddef>

// LSTM: B=64, T=512, D=1024, H=1024, gate order f,i,o,g (torch chunk order)
#define BB 64
#define TT 512
#define DD 1024
#define HH 1024
#define N4 (4 * HH)

typedef __attribute__((ext_vector_type(2))) float v2f;
typedef __attribute__((ext_vector_type(8))) float v8f;

__device__ __forceinline__ float sigmoidf_(float x) {
    return 1.0f / (1.0f + __expf(-x));
}

// ---------------------------------------------------------------------------
// Tiled transpose: in [rows][cols] -> out [cols][rows]   (both row-major f32)
// Launched with block (32,8), grid (cols/32, rows/32).
// ---------------------------------------------------------------------------
__global__ void __launch_bounds__(256)
transpose_kernel(const float* __restrict__ in, float* __restrict__ out,
                 int rows, int cols) {
    __shared__ float tile[32][33];
    const int bx = blockIdx.x * 32;   // col base in `in`
    const int by = blockIdx.y * 32;   // row base in `in`
    const int tx = threadIdx.x;
    const int ty = threadIdx.y;
#pragma unroll
    for (int i = 0; i < 32; i += 8)
        tile[ty + i][tx] = in[(size_t)(by + ty + i) * cols + (bx + tx)];
    __syncthreads();
#pragma unroll
    for (int i = 0; i < 32; i += 8)
        out[(size_t)(bx + ty + i) * rows + (by + tx)] = tile[tx][ty + i];
}

// ---------------------------------------------------------------------------
// One LSTM timestep, fused concatenated-K GEMM + gates.
//   gates[64 x 4096] = h_prev[64 x 1024] @ WhhT[1024 x 4096]
//                    + x_t   [64 x 1024] @ WihT[1024 x 4096]  (+ biases)
// Block = 128 threads (4 waves). Block tile: 16 batch rows x 16 hidden cols.
// Wave q (0..3) computes gate quadrant q (f,i,o,g) with f32 WMMA 16x16x4.
// Then LDS exchange + elementwise cell update. c updated in place (tile is
// exclusively owned); h_{t-1} read from out[:, t-1, :], h_t written to
// out[:, t, :], so no extra hidden-state buffer or races.
// ---------------------------------------------------------------------------
__global__ void __launch_bounds__(128)
lstm_step_kernel(const float* __restrict__ x,
                 const float* __restrict__ WihT,   // [DD][N4]
                 const float* __restrict__ WhhT,   // [HH][N4]
                 const float* __restrict__ b_ih,   // [N4]
                 const float* __restrict__ b_hh,   // [N4]
                 const float* __restrict__ hprev,  // [BB][...] row stride hstride
                 size_t hstride,
                 float* __restrict__ c,            // [BB][HH], in-place
                 float* __restrict__ out,          // [BB][TT][HH]
                 int t) {
    const int lane  = threadIdx.x & 31;
    const int q     = threadIdx.x >> 5;        // gate quadrant 0..3
    const int m0    = blockIdx.x * 16;         // batch tile base
    const int n0    = blockIdx.y * 16;         // hidden-col tile base
    const int row   = lane & 15;               // A: M index, B: N index
    const int khalf = (lane >> 4) * 2;         // K half-split per ISA layout

    const int ncol = q * HH + n0 + row;        // column in 4H gate space

    v8f acc = {0.f, 0.f, 0.f, 0.f, 0.f, 0.f, 0.f, 0.f};

    // ---- recurrent part: acc += h_prev @ W_hh^T ----
    {
        const float* arow = hprev + (size_t)(m0 + row) * hstride + khalf;
        const float* bp0  = WhhT + (size_t)khalf * N4 + ncol;
#pragma unroll 4
        for (int k = 0; k < HH; k += 4) {
            v2f a = *(const v2f*)(arow + k);
            const float* bp = bp0 + (size_t)k * N4;
            v2f b;
            b.x = bp[0];
            b.y = bp[N4];
            acc = __builtin_amdgcn_wmma_f32_16x16x4_f32(
                false, a, false, b, (short)0, acc, false, false);
        }
    }

    // ---- input part: acc += x_t @ W_ih^T ----
    {
        const float* arow = x + ((size_t)(m0 + row) * TT + t) * DD + khalf;
        const float* bp0  = WihT + (size_t)khalf * N4 + ncol;
#pragma unroll 4
        for (int k = 0; k < DD; k += 4) {
            v2f a = *(const v2f*)(arow + k);
            const float* bp = bp0 + (size_t)k * N4;
            v2f b;
            b.x = bp[0];
            b.y = bp[N4];
            acc = __builtin_amdgcn_wmma_f32_16x16x4_f32(
                false, a, false, b, (short)0, acc, false, false);
        }
    }

    // ---- exchange the four gate tiles through LDS ----
    __shared__ float gbuf[4][16][17];   // [quadrant][m][n], padded
#pragma unroll
    for (int r = 0; r < 8; ++r) {
        int m = r + 8 * (lane >> 4);    // C/D layout: VGPR r holds M = r, r+8
        gbuf[q][m][row] = acc[r];
    }
    __syncthreads();

    // ---- elementwise LSTM cell update (128 threads, 256 elements) ----
    for (int e = threadIdx.x; e < 256; e += 128) {
        const int m  = e >> 4;
        const int n  = e & 15;
        const int gc = n0 + n;          // hidden column

        float fv = sigmoidf_(gbuf[0][m][n] + b_ih[0 * HH + gc] + b_hh[0 * HH + gc]);
        float iv = sigmoidf_(gbuf[1][m][n] + b_ih[1 * HH + gc] + b_hh[1 * HH + gc]);
        float ov = sigmoidf_(gbuf[2][m][n] + b_ih[2 * HH + gc] + b_hh[2 * HH + gc]);
        float gv = tanhf    (gbuf[3][m][n] + b_ih[3 * HH + gc] + b_hh[3 * HH + gc]);

        const size_t cidx = (size_t)(m0 + m) * HH + gc;
        const float cn = fv * c[cidx] + iv * gv;
        const float hn = ov * tanhf(cn);
        c[cidx] = cn;
        out[((size_t)(m0 + m) * TT + t) * HH + gc] = hn;
    }
}

// ---------------------------------------------------------------------------
extern "C" void kernel_launch(void* const* d_in, const int* in_sizes, int n_in,
                              void* d_out, int out_size, void* d_ws, size_t ws_size,
                              hipStream_t stream) {
    const float* x    = (const float*)d_in[0];   // [BB][TT][DD]
    const float* W_ih = (const float*)d_in[1];   // [N4][DD]
    const float* W_hh = (const float*)d_in[2];   // [N4][HH]
    const float* b_ih = (const float*)d_in[3];   // [N4]
    const float* b_hh = (const float*)d_in[4];   // [N4]
    float* out = (float*)d_out;                  // [BB][TT][HH]

    // workspace layout (~34 MB)
    float* WihT = (float*)d_ws;                  // [DD][N4]
    float* WhhT = WihT + (size_t)DD * N4;        // [HH][N4]
    float* cst  = WhhT + (size_t)HH * N4;        // [BB][HH] cell state
    float* h0   = cst  + (size_t)BB * HH;        // [BB][HH] zero initial h

    // zero c and h0 (contiguous) every call — deterministic, graph-safe
    hipMemsetAsync(cst, 0, (size_t)2 * BB * HH * sizeof(float), stream);

    // one-time transposes so WMMA B-fragment loads are N-contiguous
    dim3 tb(32, 8);
    transpose_kernel<<<dim3(DD / 32, N4 / 32), tb, 0, stream>>>(W_ih, WihT, N4, DD);
    transpose_kernel<<<dim3(HH / 32, N4 / 32), tb, 0, stream>>>(W_hh, WhhT, N4, HH);

    // sequential scan over time; h_{t-1} comes straight from the output tensor
    for (int t = 0; t < TT; ++t) {
        const float* hprev = t ? (out + (size_t)(t - 1) * HH) : h0;
        size_t hstride     = t ? (size_t)TT * HH : (size_t)HH;
        lstm_step_kernel<<<dim3(BB / 16, HH / 16), 128, 0, stream>>>(
            x, WihT, WhhT, b_ih, b_hh, hprev, hstride, cst, out, t);
    }
}